// fullNN_40973988004614
// MI455X (gfx1250) — compile-verified
//
#include <hip/hip_runtime.h>

// fullNN SDE+BN-MLP scan for MI455X (gfx1250, wave32).
// BN∘Linear is affine => collapse each layer to affine maps whose coefficients
// come from exact batch moments. 3 reduction passes per step.
// All batch-moment accumulations (Gram matrices of augmented feature vectors)
// and the layer-2 GEMM use V_WMMA_F32_16X16X4_F32 in full fp32.
// W+S (128MB) + ~17MB scratch stay L2-resident (192MB L2).

#define T_STEPS 64
#define NSCAN   63
#define WG      256
#define DT_     (1.0f/64.0f)
#define MU_V_   0.05f
#define MU_D_   0.01f
#define SIG_V_  0.2f
#define SIG_D_  0.1f
#define R_      0.05f
#define EPS_    1e-6f

typedef __attribute__((ext_vector_type(2))) float v2f;
typedef __attribute__((ext_vector_type(8))) float v8f;

// COEF layout (floats):
// [0] z0  [1] z1  [2] h3(path0)  [3] h3(path1)
// [4..47]   A1p[f*11+j] (layer-1 folded affine, 4x11)
// [48..58]  c1p[j]
// [64..255] A2p (12x16, zero padded; row-major [k*16+n])  -- WMMA B matrix
// [256..271] c2p[n] (zero padded)
// [272..287] w2p[n] = W2 column (zero padded)
// [288] bias2 scalar

// ds_swizzle XOR-butterfly (group-of-32 mode: and=0x1f, or=0, xor=k<<10).
template <int IMM>
__device__ __forceinline__ float dswz(float v) {
  return __int_as_float(__builtin_amdgcn_ds_swizzle(__float_as_int(v), IMM));
}

__global__ void k_init(float4* __restrict__ ST, float* __restrict__ COEF,
                       const float* __restrict__ y0i, const float* __restrict__ y1i,
                       const float* __restrict__ z0i, const float* __restrict__ z1i) {
  int b = blockIdx.x * WG + threadIdx.x;
  ST[b] = make_float4(1.0f, 0.0f, y0i[0], y1i[0]);
  if (b == 0) { COEF[0] = z0i[0]; COEF[1] = z1i[0]; COEF[2] = 0.0f; COEF[3] = 0.0f; }
}

// Pass 1: SDE update; accumulate moments of the 4-dim state via a WMMA Gram of
// the augmented vector (x0,x1,y0,y1,1): Gram row 4 = sums, Gram[i][j] = products.
// Per 4 paths, A[m][k]=X[k][m] and B[k][n]=X[k][n] share the same lane/VGPR
// layout (lane half = K/2, vgpr = K&1), so one v2f feeds both operands.
__global__ void k_pass1(const float* __restrict__ Wb, const float* __restrict__ Sb,
                        float4* __restrict__ ST, float* __restrict__ PART,
                        const float* __restrict__ COEF, int t) {
  int b = blockIdx.x * WG + threadIdx.x;
  float z0 = COEF[0], z1 = COEF[1];
  float w0 = Wb[(size_t)b * T_STEPS + t];
  float w1 = Wb[(size_t)b * T_STEPS + t + 1];
  float s  = Sb[(size_t)b * T_STEPS + t];
  float dw = w1 - w0;
  float4 st = ST[b];
  float x0 = st.x, x1 = st.y, y0 = st.z, y1 = st.w;
  float x0n = x0 + (MU_V_ * s * x0 + MU_D_ * s * x1) * DT_
                 + (SIG_V_ * s * x0 + SIG_D_ * s * x1) * dw;
  float y0n = y0 + R_ * y0 * DT_ + z0 * dw;
  float y1n = y1 + R_ * y1 * DT_ + z1 * dw;
  float x1n = x1 - y1n * DT_;
  ST[b] = make_float4(x0n, x1n, y0n, y1n);

  __shared__ float blk[WG * 16];          // per-path feature row, stride 16
  __shared__ float red[8][25];
  int wv = threadIdx.x >> 5, ln = threadIdx.x & 31;
  int half = ln >> 4, n16 = ln & 15;

  float4* myrow = (float4*)(blk + (threadIdx.x << 4));
  myrow[0] = make_float4(x0n, x1n, y0n, y1n);
  myrow[1] = make_float4(1.0f, 0.0f, 0.0f, 0.0f);
  myrow[2] = make_float4(0.0f, 0.0f, 0.0f, 0.0f);
  myrow[3] = make_float4(0.0f, 0.0f, 0.0f, 0.0f);
  __syncthreads();

  const float* wb = blk + (wv << 9);      // this wave's 32x16 block
  v8f C = {};
#pragma unroll
  for (int c = 0; c < 8; ++c) {
    int k0 = c * 4 + half * 2;
    v2f AB;
    AB.x = wb[k0 * 16 + n16];
    AB.y = wb[(k0 + 1) * 16 + n16];
    C = __builtin_amdgcn_wmma_f32_16x16x4_f32(false, AB, false, AB, (short)0, C, false, false);
  }
  // C[v] = Gram[m][n], m = v + 8*half, n = n16 (valid m,n <= 4).
  if (half == 0 && n16 < 5) {
#pragma unroll
    for (int v = 0; v < 5; ++v) red[wv][v * 5 + n16] = C[v];
  }
  __syncthreads();
  if (threadIdx.x < 14) {
    int idx = threadIdx.x, i, j;
    if (idx < 4) { i = 4; j = idx; }         // sums from augmented row
    else { int r = idx - 4; i = 0; while (r >= 4 - i) { r -= 4 - i; ++i; } j = i + r; }
    float a = 0.0f;
#pragma unroll
    for (int w = 0; w < 8; ++w) a += red[w][i * 5 + j];   // fixed order: deterministic
    PART[(size_t)blockIdx.x * 80 + threadIdx.x] = a;
  }
}

// Small 1: reduce 14 sums over all workgroups; build folded layer-1 affine.
__global__ void k_small1(const float* __restrict__ PART, float* __restrict__ COEF,
                         const float* __restrict__ g0, const float* __restrict__ b0,
                         const float* __restrict__ W0, const float* __restrict__ g1,
                         const float* __restrict__ b1, int t, int nwg, float invB) {
  __shared__ float S[14];
  int tid = threadIdx.x;
  if (tid < 14) {
    float a = 0.0f;
    for (int w = 0; w < nwg; ++w) a += PART[(size_t)w * 80 + tid];
    S[tid] = a;
  }
  __syncthreads();
  if (tid == 0) {
    float mu[4], C[4][4];
    for (int f = 0; f < 4; ++f) mu[f] = S[f] * invB;
    int k = 4;
    for (int i = 0; i < 4; ++i)
      for (int j = i; j < 4; ++j) {
        float c = S[k] * invB - mu[i] * mu[j];
        C[i][j] = c; C[j][i] = c; ++k;
      }
    float s0[4], t0[4];
    for (int f = 0; f < 4; ++f) {
      float sf = g0[t * 4 + f] * rsqrtf(C[f][f] + EPS_);
      s0[f] = sf;
      t0[f] = b0[t * 4 + f] - mu[f] * sf;
    }
    for (int j = 0; j < 11; ++j) {
      float a[4], c1 = 0.0f, m1 = 0.0f;
      for (int f = 0; f < 4; ++f) {
        float w = W0[(size_t)t * 44 + f * 11 + j];
        a[f] = s0[f] * w;
        c1 += t0[f] * w;
        m1 += mu[f] * a[f];
      }
      m1 += c1;
      float v1 = 0.0f;
      for (int f = 0; f < 4; ++f) {
        float acc = 0.0f;
        for (int g = 0; g < 4; ++g) acc += C[f][g] * a[g];
        v1 += a[f] * acc;
      }
      float s1 = g1[t * 11 + j] * rsqrtf(v1 + EPS_);
      float t1 = b1[t * 11 + j] - m1 * s1;
      for (int f = 0; f < 4; ++f) COEF[4 + f * 11 + j] = a[f] * s1;
      COEF[48 + j] = c1 * s1 + t1;
    }
  }
}

// Pass 2: r1 = relu(state @ A1p + c1p); store (Bx12) for pass 3; accumulate
// mean(11) + Gram(66) of r1 with WMMA on the augmented vector (r1, 1).
__global__ void k_pass2(const float4* __restrict__ ST, float* __restrict__ R1,
                        float* __restrict__ PART, const float* __restrict__ COEF) {
  int b = blockIdx.x * WG + threadIdx.x;
  float4 st = ST[b];
  float z[4] = {st.x, st.y, st.z, st.w};
  float r1[11];
#pragma unroll
  for (int j = 0; j < 11; ++j) {
    float h = COEF[48 + j];
#pragma unroll
    for (int f = 0; f < 4; ++f) h = fmaf(z[f], COEF[4 + f * 11 + j], h);
    r1[j] = fmaxf(h, 0.0f);
  }
  float4* rp = (float4*)(R1 + (size_t)b * 12);
  rp[0] = make_float4(r1[0], r1[1], r1[2], r1[3]);
  rp[1] = make_float4(r1[4], r1[5], r1[6], r1[7]);
  rp[2] = make_float4(r1[8], r1[9], r1[10], 0.0f);    // K-pad for pass-3 WMMA

  __shared__ float blk[WG * 16];
  __shared__ float red[8][144];
  int wv = threadIdx.x >> 5, ln = threadIdx.x & 31;
  int half = ln >> 4, n16 = ln & 15;

  float4* myrow = (float4*)(blk + (threadIdx.x << 4));
  myrow[0] = make_float4(r1[0], r1[1], r1[2], r1[3]);
  myrow[1] = make_float4(r1[4], r1[5], r1[6], r1[7]);
  myrow[2] = make_float4(r1[8], r1[9], r1[10], 1.0f);  // augmented 1 => row 11 = sums
  myrow[3] = make_float4(0.0f, 0.0f, 0.0f, 0.0f);
  __syncthreads();

  const float* wb = blk + (wv << 9);
  v8f C = {};
#pragma unroll
  for (int c = 0; c < 8; ++c) {
    int k0 = c * 4 + half * 2;
    v2f AB;
    AB.x = wb[k0 * 16 + n16];
    AB.y = wb[(k0 + 1) * 16 + n16];
    C = __builtin_amdgcn_wmma_f32_16x16x4_f32(false, AB, false, AB, (short)0, C, false, false);
  }
  // C[v] = Gram12[m][n], m = v + 8*half, n = n16 (valid m,n <= 11).
  if (n16 < 12) {
#pragma unroll
    for (int v = 0; v < 8; ++v) {
      int m = v + 8 * half;
      if (m < 12) red[wv][m * 12 + n16] = C[v];
    }
  }
  __syncthreads();
  if (threadIdx.x < 77) {
    int idx = threadIdx.x, i, j;
    if (idx < 11) { i = 11; j = idx; }       // sums from augmented row
    else { int r = idx - 11; i = 0; while (r >= 11 - i) { r -= 11 - i; ++i; } j = i + r; }
    float a = 0.0f;
#pragma unroll
    for (int w = 0; w < 8; ++w) a += red[w][i * 12 + j];
    PART[(size_t)blockIdx.x * 80 + threadIdx.x] = a;
  }
}

// Small 2: reduce 77 sums; build folded layer-2 affine as the WMMA B matrix.
__global__ void k_small2(const float* __restrict__ PART, float* __restrict__ COEF,
                         const float* __restrict__ W1, const float* __restrict__ g2,
                         const float* __restrict__ b2, const float* __restrict__ W2,
                         const float* __restrict__ bias2, int t, int nwg, float invB) {
  __shared__ float S[77];
  int tid = threadIdx.x;
  if (tid < 77) {
    float a = 0.0f;
    for (int w = 0; w < nwg; ++w) a += PART[(size_t)w * 80 + tid];
    S[tid] = a;
  }
  __syncthreads();
  if (tid < 16) {
    int j = tid;
    if (j < 11) {
      float wcol[11], mean[11];
      for (int i = 0; i < 11; ++i) {
        wcol[i] = W1[(size_t)t * 121 + i * 11 + j];
        mean[i] = S[i] * invB;
      }
      float m2 = 0.0f;
      for (int i = 0; i < 11; ++i) m2 = fmaf(mean[i], wcol[i], m2);
      float v2 = 0.0f;
      for (int i = 0; i < 11; ++i)
        for (int kk = 0; kk < 11; ++kk) {
          int a2 = i < kk ? i : kk, c2i = i < kk ? kk : i;
          int idx = 11 + a2 * 11 - (a2 * (a2 - 1)) / 2 + (c2i - a2);
          float cov = S[idx] * invB - mean[i] * mean[kk];
          v2 = fmaf(wcol[i] * cov, wcol[kk], v2);
        }
      float s2 = g2[t * 11 + j] * rsqrtf(v2 + EPS_);
      float c2 = b2[t * 11 + j] - m2 * s2;
      for (int i = 0; i < 11; ++i) COEF[64 + i * 16 + j] = wcol[i] * s2;
      COEF[64 + 11 * 16 + j] = 0.0f;  // pad K row 11
      COEF[256 + j] = c2;
      COEF[272 + j] = W2[t * 11 + j];
    } else {
      for (int i = 0; i < 12; ++i) COEF[64 + i * 16 + j] = 0.0f;
      COEF[256 + j] = 0.0f;
      COEF[272 + j] = 0.0f;
    }
    if (j == 0) COEF[288] = bias2[t];
  }
}

// Pass 3: H2 = R1 @ A2p via 3 chained V_WMMA_F32_16X16X4_F32 per 16-path tile;
// r2 = relu(H2 + c2p); h3 = r2.w2 + bias2 (w2 zero-padded => no N masking).
// Row sum over N uses immediate ds_swizzle XOR butterflies (no bpermute/cmp).
__global__ void k_pass3(const float* __restrict__ R1, float* __restrict__ PART,
                        float* __restrict__ COEF) {
  int wv = threadIdx.x >> 5, ln = threadIdx.x & 31;
  int half = ln >> 4, n = ln & 15;

  v2f Bm0, Bm1, Bm2;
  {
    int r0 = half * 2;
    Bm0.x = COEF[64 + (r0 + 0) * 16 + n];  Bm0.y = COEF[64 + (r0 + 1) * 16 + n];
    Bm1.x = COEF[64 + (r0 + 4) * 16 + n];  Bm1.y = COEF[64 + (r0 + 5) * 16 + n];
    Bm2.x = COEF[64 + (r0 + 8) * 16 + n];  Bm2.y = COEF[64 + (r0 + 9) * 16 + n];
  }
  float c2v = COEF[256 + n];
  float w2v = COEF[272 + n];
  float b2s = COEF[288];

  float acc1 = 0.0f, acc2 = 0.0f;
  int base = blockIdx.x * 256 + wv * 32;
#pragma unroll
  for (int tile = 0; tile < 2; ++tile) {
    int b0 = base + tile * 16;
    const float* rrow = R1 + (size_t)(b0 + n) * 12 + half * 2;
    v8f C = {};
    v2f A;
    A.x = rrow[0]; A.y = rrow[1];
    C = __builtin_amdgcn_wmma_f32_16x16x4_f32(false, A, false, Bm0, (short)0, C, false, false);
    A.x = rrow[4]; A.y = rrow[5];
    C = __builtin_amdgcn_wmma_f32_16x16x4_f32(false, A, false, Bm1, (short)0, C, false, false);
    A.x = rrow[8]; A.y = rrow[9];
    C = __builtin_amdgcn_wmma_f32_16x16x4_f32(false, A, false, Bm2, (short)0, C, false, false);

    float h3[8];
#pragma unroll
    for (int v = 0; v < 8; ++v) {
      float x = fmaxf(C[v] + c2v, 0.0f) * w2v;
      x += dswz<0x041F>(x);   // xor 1
      x += dswz<0x081F>(x);   // xor 2
      x += dswz<0x101F>(x);   // xor 4
      x += dswz<0x201F>(x);   // xor 8
      h3[v] = x + b2s;
    }
    float ls1 = 0.0f, ls2 = 0.0f;
#pragma unroll
    for (int v = 0; v < 8; ++v) { ls1 += h3[v]; ls2 = fmaf(h3[v], h3[v], ls2); }
    ls1 += dswz<0x401F>(ls1);   // combine lane halves (rows 0-7 + rows 8-15)
    ls2 += dswz<0x401F>(ls2);
    acc1 += ls1; acc2 += ls2;

    if (b0 == 0 && ln == 0) { COEF[2] = h3[0]; COEF[3] = h3[1]; }
  }

  __shared__ float red[8][2];
  if (!ln) { red[wv][0] = acc1; red[wv][1] = acc2; }
  __syncthreads();
  if (threadIdx.x < 2) {
    float a = 0.0f;
#pragma unroll
    for (int w = 0; w < 8; ++w) a += red[w][threadIdx.x];
    PART[(size_t)blockIdx.x * 80 + threadIdx.x] = a;
  }
}

// Small 3: m3/v3 from the two sums; z0,z1 = BN3 of paths 0,1.
__global__ void k_small3(const float* __restrict__ PART, float* __restrict__ COEF,
                         const float* __restrict__ g3, const float* __restrict__ b3,
                         int t, int nwg, float invB) {
  __shared__ float S[2];
  int tid = threadIdx.x;
  if (tid < 2) {
    float a = 0.0f;
    for (int w = 0; w < nwg; ++w) a += PART[(size_t)w * 80 + tid];
    S[tid] = a;
  }
  __syncthreads();
  if (tid == 0) {
    float m3 = S[0] * invB;
    float v3 = S[1] * invB - m3 * m3;
    float sc = g3[t] * rsqrtf(v3 + EPS_);
    COEF[0] = (COEF[2] - m3) * sc + b3[t];
    COEF[1] = (COEF[3] - m3) * sc + b3[t];
  }
}

// Final step: note y1 uses -dw per the reference epilogue.
__global__ void k_final(const float* __restrict__ Wb, const float* __restrict__ Sb,
                        const float4* __restrict__ ST, const float* __restrict__ COEF,
                        float* __restrict__ out, int nB) {
  int b = blockIdx.x * WG + threadIdx.x;
  float z0 = COEF[0], z1 = COEF[1];
  float dw = Wb[(size_t)b * T_STEPS + 63] - Wb[(size_t)b * T_STEPS + 62];
  float s  = Sb[(size_t)b * T_STEPS + 63];
  float4 st = ST[b];
  float x0 = st.x, x1 = st.y, y0 = st.z, y1 = st.w;
  float x0n = x0 + (MU_V_ * s * x0 + MU_D_ * s * x1) * DT_
                 + (SIG_V_ * s * x0 + SIG_D_ * s * x1) * dw;
  float y0n = y0 + R_ * y0 * DT_ + z0 * dw;
  float y1n = y1 + R_ * y1 * DT_ - z1 * dw;
  float x1n = x1 - y1n * DT_;
  out[b] = x0n;
  out[(size_t)nB + b] = x1n;
  out[(size_t)2 * nB + b] = y0n;
  out[(size_t)3 * nB + b] = y1n;
}

extern "C" void kernel_launch(void* const* d_in, const int* in_sizes, int n_in,
                              void* d_out, int out_size, void* d_ws, size_t ws_size,
                              hipStream_t stream) {
  (void)n_in; (void)out_size; (void)ws_size;
  const float* W   = (const float*)d_in[0];
  const float* S   = (const float*)d_in[1];
  const float* y0i = (const float*)d_in[2];
  const float* y1i = (const float*)d_in[3];
  const float* z0i = (const float*)d_in[4];
  const float* z1i = (const float*)d_in[5];
  const float* g0  = (const float*)d_in[6];
  const float* b0  = (const float*)d_in[7];
  const float* W0  = (const float*)d_in[8];
  const float* g1  = (const float*)d_in[9];
  const float* b1  = (const float*)d_in[10];
  const float* W1  = (const float*)d_in[11];
  const float* g2  = (const float*)d_in[12];
  const float* b2  = (const float*)d_in[13];
  const float* W2  = (const float*)d_in[14];
  const float* bi2 = (const float*)d_in[15];
  const float* g3  = (const float*)d_in[16];
  const float* b3  = (const float*)d_in[17];

  const int nB  = in_sizes[0] / T_STEPS;   // 262144
  const int nwg = nB / WG;                 // 1024
  const float invB = 1.0f / (float)nB;

  char* ws = (char*)d_ws;
  float4* ST  = (float4*)ws;                                            // 16B * nB
  float* R1   = (float*)(ws + (size_t)nB * 16);                         // 48B * nB
  float* PART = (float*)(ws + (size_t)nB * 64);                         // 320B * nwg
  float* COEF = (float*)(ws + (size_t)nB * 64 + (size_t)nwg * 320);     // 512 floats

  dim3 big(nwg), one(1), blk(WG);
  k_init<<<big, blk, 0, stream>>>(ST, COEF, y0i, y1i, z0i, z1i);
  for (int t = 0; t < NSCAN; ++t) {
    k_pass1<<<big, blk, 0, stream>>>(W, S, ST, PART, COEF, t);
    k_small1<<<one, blk, 0, stream>>>(PART, COEF, g0, b0, W0, g1, b1, t, nwg, invB);
    k_pass2<<<big, blk, 0, stream>>>(ST, R1, PART, COEF);
    k_small2<<<one, blk, 0, stream>>>(PART, COEF, W1, g2, b2, W2, bi2, t, nwg, invB);
    k_pass3<<<big, blk, 0, stream>>>(R1, PART, COEF);
    k_small3<<<one, blk, 0, stream>>>(PART, COEF, g3, b3, t, nwg, invB);
  }
  k_final<<<big, blk, 0, stream>>>(W, S, ST, COEF, (float*)d_out, nB);
}